// TEAVARParameterModel_39006892982859
// MI455X (gfx1250) — compile-verified
//
#include <hip/hip_runtime.h>
#include <math.h>

// ---------------- problem constants ----------------
#define BD     8      // batch
#define DD     132    // demands
#define TPD    4
#define TT     528    // tunnels
#define SS     32     // scenarios
#define EE     48     // edges
#define HIST   50
#define INDIM  6600   // D*HIST
#define HID    4096
#define NWRD   17     // ceil(528/32)
#define RHO_   10.0f
#define LR_    1e-3f
#define ITERS_ 200

typedef float v2f __attribute__((ext_vector_type(2)));
typedef float v8f __attribute__((ext_vector_type(8)));

// ---------------------------------------------------------------------------
// Kernel 1: base_demand stats (mean + unbiased std) -> tmWS [B,D]
// ---------------------------------------------------------------------------
__global__ __launch_bounds__(256)
void teavar_stats(const float* __restrict__ hist, float* __restrict__ tmWS) {
    int idx = blockIdx.x * blockDim.x + threadIdx.x;
    if (idx >= BD * DD) return;
    int b = idx / DD, d = idx - b * DD;
    const float* p = hist + (size_t)b * INDIM + d * HIST;
    float s = 0.f;
    #pragma unroll 10
    for (int i = 0; i < HIST; ++i) s += p[i];
    float mean = s * (1.0f / HIST);
    float ss = 0.f;
    #pragma unroll 10
    for (int i = 0; i < HIST; ++i) { float t = p[i] - mean; ss += t * t; }
    float sd = sqrtf(ss * (1.0f / (HIST - 1)));
    tmWS[idx] = mean * (1.0f + 0.3f * sd / (mean + 1e-6f));
}

// ---------------------------------------------------------------------------
// Kernel 2: pack masks.
//   eb    [EE*NWRD] : edge-major tunnel bitmask (for g2 dot products)
//   aliveT[TT]      : per-tunnel scenario bitmask (bit s)
//   edgeT [TT*2]    : per-tunnel edge bitmask (48 bits in 2 words)
// ---------------------------------------------------------------------------
__global__ __launch_bounds__(256)
void teavar_pack(const float* __restrict__ alive, const float* __restrict__ einc,
                 unsigned* __restrict__ eb, unsigned* __restrict__ aliveT,
                 unsigned* __restrict__ edgeT) {
    int idx = blockIdx.x * blockDim.x + threadIdx.x;
    if (idx < EE * NWRD) {
        int e = idx / NWRD, w = idx - e * NWRD;
        unsigned m = 0;
        for (int j = 0; j < 32; ++j) {
            int t = w * 32 + j;
            if (t < TT && einc[e * TT + t] > 0.5f) m |= (1u << j);
        }
        eb[idx] = m;
    } else if (idx < EE * NWRD + TT) {
        int t = idx - EE * NWRD;
        unsigned m = 0;
        for (int s = 0; s < SS; ++s)
            if (alive[s * TT + t] > 0.5f) m |= (1u << s);
        aliveT[t] = m;
    } else if (idx < EE * NWRD + 2 * TT) {
        int t = idx - EE * NWRD - TT;
        unsigned m0 = 0, m1 = 0;
        for (int e = 0; e < 32; ++e)
            if (einc[e * TT + t] > 0.5f) m0 |= (1u << e);
        for (int e = 32; e < EE; ++e)
            if (einc[e * TT + t] > 0.5f) m1 |= (1u << (e - 32));
        edgeT[2 * t]     = m0;
        edgeT[2 * t + 1] = m1;
    }
}

// ---------------------------------------------------------------------------
// Kernel 3: h[16,4096] = relu(Apad[16,6600] @ W1[6600,4096] + b1)
// One wave per 16-col tile; straight-line pipelined V_WMMA_F32_16X16X4_F32.
// A rows 8..15 read clamped row (l&7) and are zeroed by a lane mask (no EXEC
// divergence). W1 streamed once -> non-temporal loads.
// ---------------------------------------------------------------------------
__global__ __launch_bounds__(128)
void teavar_gemm1(const float* __restrict__ hist, const float* __restrict__ W1,
                  const float* __restrict__ b1, float* __restrict__ hOut) {
    const int lane = threadIdx.x & 31;
    const int wave = threadIdx.x >> 5;
    const int tile = blockIdx.x * 4 + wave;        // 0..255
    const int col0 = tile * 16;
    const int half = lane >> 4;                    // lanes 16-31 carry K+2/K+3
    const int l    = lane & 15;                    // M row (A) / N col (B)
    const int klo  = half * 2;
    const int coln = col0 + l;

    const float amask = (l < BD) ? 1.0f : 0.0f;    // zero out padded rows
    const float* aBase = hist + (size_t)(l & 7) * INDIM;
    const float* bBase = W1 + coln;                // B[k,n] row-major, pitch 4096

    v8f acc = {};
    v2f a  = *(const v2f*)(aBase + klo);
    v2f bb = { __builtin_nontemporal_load(bBase + (size_t)klo * HID),
               __builtin_nontemporal_load(bBase + (size_t)(klo + 1) * HID) };
    #pragma unroll 8
    for (int kk = 0; kk < INDIM - 4; kk += 4) {
        const int kn = kk + 4 + klo;
        v2f a_n = *(const v2f*)(aBase + kn);
        v2f b_n = { __builtin_nontemporal_load(bBase + (size_t)kn * HID),
                    __builtin_nontemporal_load(bBase + (size_t)(kn + 1) * HID) };
        v2f am = { a.x * amask, a.y * amask };
        acc = __builtin_amdgcn_wmma_f32_16x16x4_f32(
                false, am, false, bb, (short)0, acc, false, false);
        a = a_n; bb = b_n;
    }
    {
        v2f am = { a.x * amask, a.y * amask };
        acc = __builtin_amdgcn_wmma_f32_16x16x4_f32(
                false, am, false, bb, (short)0, acc, false, false);
    }
    float bias = b1[coln];
    #pragma unroll
    for (int v = 0; v < 8; ++v) {
        int row = half * 8 + v;                    // C layout: vgpr v = row v / 8+v
        float c = acc[v] + bias;
        hOut[(size_t)row * HID + coln] = c > 0.f ? c : 0.f;
    }
}

// ---------------------------------------------------------------------------
// Kernel 4: edge_theta = softplus(h[16,4096] @ W2[4096,48] + b2); cap = base*theta
// ---------------------------------------------------------------------------
__global__ __launch_bounds__(96)
void teavar_gemm2(const float* __restrict__ h, const float* __restrict__ W2,
                  const float* __restrict__ b2, const float* __restrict__ base_cap,
                  float* __restrict__ thetaOut, float* __restrict__ capWS) {
    const int lane = threadIdx.x & 31;
    const int tile = threadIdx.x >> 5;             // 0..2
    const int col0 = tile * 16;
    const int half = lane >> 4;
    const int l    = lane & 15;
    const int klo  = half * 2;
    const int coln = col0 + l;

    const float* aBase = h + (size_t)l * HID;
    const float* bBase = W2 + coln;                // pitch 48

    v8f acc = {};
    v2f a  = *(const v2f*)(aBase + klo);
    v2f bb = { bBase[(size_t)klo * EE], bBase[(size_t)(klo + 1) * EE] };
    #pragma unroll 8
    for (int kk = 0; kk < HID - 4; kk += 4) {
        const int kn = kk + 4 + klo;
        v2f a_n = *(const v2f*)(aBase + kn);
        v2f b_n = { bBase[(size_t)kn * EE], bBase[(size_t)(kn + 1) * EE] };
        acc = __builtin_amdgcn_wmma_f32_16x16x4_f32(
                false, a, false, bb, (short)0, acc, false, false);
        a = a_n; bb = b_n;
    }
    acc = __builtin_amdgcn_wmma_f32_16x16x4_f32(
            false, a, false, bb, (short)0, acc, false, false);

    if (half == 0) {                               // rows 0..7 = real batch rows
        float bias = b2[coln];
        float bc   = base_cap[coln];
        #pragma unroll
        for (int v = 0; v < 8; ++v) {
            float z  = acc[v] + bias;
            float sp = (z > 20.f) ? z : log1pf(expf(z));   // softplus
            thetaOut[v * EE + coln] = sp;
            capWS[v * EE + coln]    = bc * sp;
        }
    }
}

// ---------------------------------------------------------------------------
// Kernel 5: LP penalty projected-gradient, 200 iters, one workgroup per batch.
// All state in LDS; masks iterated with ctz pop-loops.
// ---------------------------------------------------------------------------
__global__ __launch_bounds__(256)
void teavar_lp(const float* __restrict__ tmWS, const float* __restrict__ capWS,
               const unsigned* __restrict__ ebG, const unsigned* __restrict__ aliveTG,
               const unsigned* __restrict__ edgeTG, const float* __restrict__ prob,
               float* __restrict__ xOut) {
    __shared__ float    x[TT];
    __shared__ float    u[SS];
    __shared__ float    g1[SS * DD];
    __shared__ float    g2[EE];
    __shared__ float    tm[DD];
    __shared__ float    cap[EE];
    __shared__ float    probw[SS];
    __shared__ unsigned eb[EE * NWRD];
    __shared__ unsigned aliveT[TT];
    __shared__ unsigned edgeT[2 * TT];
    __shared__ float    red[256];
    __shared__ float    alphaS;

    const int b = blockIdx.x, tid = threadIdx.x;
    for (int i = tid; i < TT; i += 256) { x[i] = 0.f; aliveT[i] = aliveTG[i]; }
    for (int i = tid; i < SS; i += 256) { u[i] = 0.f; probw[i] = prob[i] * (1.0f / (1.0f - 0.999f)); }
    for (int i = tid; i < DD; i += 256) tm[i]  = tmWS[b * DD + i];
    for (int i = tid; i < EE; i += 256) cap[i] = capWS[b * EE + i];
    for (int i = tid; i < EE * NWRD; i += 256) eb[i] = ebG[i];
    for (int i = tid; i < 2 * TT; i += 256) edgeT[i] = edgeTG[i];
    if (tid == 0) alphaS = 1.f;
    __syncthreads();

    for (int it = 0; it < ITERS_; ++it) {
        const float alpha = alphaS;
        // ---- phase A: g1[s,d] and g2[e] (read old x/u/alpha only) ----
        for (int idx = tid; idx < SS * DD; idx += 256) {
            int s = idx / DD, d = idx - s * DD;
            int t0 = d * TPD;
            float flow = 0.f;
            if ((aliveT[t0]     >> s) & 1u) flow += x[t0];
            if ((aliveT[t0 + 1] >> s) & 1u) flow += x[t0 + 1];
            if ((aliveT[t0 + 2] >> s) & 1u) flow += x[t0 + 2];
            if ((aliveT[t0 + 3] >> s) & 1u) flow += x[t0 + 3];
            float g = (1.f - alpha - u[s]) * tm[d] - flow;
            g1[idx] = g > 0.f ? g : 0.f;
        }
        if (tid < EE) {
            float acc = 0.f;
            for (int w = 0; w < NWRD; ++w) {
                unsigned m = eb[tid * NWRD + w];
                while (m) {
                    int j = __builtin_ctz(m);
                    m &= m - 1;
                    acc += x[w * 32 + j];
                }
            }
            float g = acc - cap[tid];
            g2[tid] = g > 0.f ? g : 0.f;
        }
        __syncthreads();

        // ---- phase B: gradients & projected updates ----
        float part = 0.f;                           // alpha partial: sum g1*tm
        for (int idx = tid; idx < SS * DD; idx += 256) {
            int d = idx % DD;
            part += g1[idx] * tm[d];
        }
        red[tid] = part;

        for (int t = tid; t < TT; t += 256) {
            int d = t >> 2;
            float sumS = 0.f;
            unsigned ms = aliveT[t];
            while (ms) {
                int s = __builtin_ctz(ms);
                ms &= ms - 1;
                sumS += g1[s * DD + d];
            }
            float sumE = 0.f;
            unsigned m0 = edgeT[2 * t], m1 = edgeT[2 * t + 1];
            while (m0) { int e = __builtin_ctz(m0); m0 &= m0 - 1; sumE += g2[e]; }
            while (m1) { int e = __builtin_ctz(m1); m1 &= m1 - 1; sumE += g2[32 + e]; }
            float xn = x[t] - LR_ * (2.f * RHO_ * (sumE - sumS));
            x[t] = xn > 0.f ? xn : 0.f;
        }
        if (tid < SS) {
            float sd = 0.f;
            for (int d = 0; d < DD; ++d) sd += g1[tid * DD + d] * tm[d];
            float un = u[tid] - LR_ * (probw[tid] - 2.f * RHO_ * sd);
            u[tid] = un > 0.f ? un : 0.f;
        }
        __syncthreads();
        if (tid == 0) {
            float tot = 0.f;
            for (int i = 0; i < 256; ++i) tot += red[i];
            float an = alphaS - LR_ * (1.f - 2.f * RHO_ * tot);
            alphaS = an > 0.f ? an : 0.f;
        }
        __syncthreads();
    }
    for (int t = tid; t < TT; t += 256) xOut[b * TT + t] = x[t];
}

// ---------------------------------------------------------------------------
extern "C" void kernel_launch(void* const* d_in, const int* in_sizes, int n_in,
                              void* d_out, int out_size, void* d_ws, size_t ws_size,
                              hipStream_t stream) {
    const float* hist     = (const float*)d_in[0];   // [8,6600]
    const float* W1       = (const float*)d_in[1];   // [6600,4096]
    const float* b1       = (const float*)d_in[2];   // [4096]
    const float* W2       = (const float*)d_in[3];   // [4096,48]
    const float* b2       = (const float*)d_in[4];   // [48]
    const float* base_cap = (const float*)d_in[5];   // [48]
    // d_in[6] = A (implicit: demand = tunnel/4)
    const float* alive    = (const float*)d_in[7];   // [32,528]
    const float* einc     = (const float*)d_in[8];   // [48,528]
    const float* prob     = (const float*)d_in[9];   // [32]

    float* xOut     = (float*)d_out;                 // [8,528]
    float* thetaOut = xOut + BD * TT;                // [8,48]

    float*    hWS    = (float*)d_ws;                 // [16,4096]
    float*    tmWS   = hWS + 16 * HID;               // [8,132]
    float*    capWS  = tmWS + BD * DD;               // [8,48]
    unsigned* ebWS   = (unsigned*)(capWS + BD * EE); // [48*17]
    unsigned* aliveT = ebWS + EE * NWRD;             // [528]
    unsigned* edgeT  = aliveT + TT;                  // [2*528]

    teavar_stats<<<(BD * DD + 255) / 256, 256, 0, stream>>>(hist, tmWS);
    teavar_pack <<<(EE * NWRD + 2 * TT + 255) / 256, 256, 0, stream>>>(alive, einc, ebWS, aliveT, edgeT);
    teavar_gemm1<<<64, 128, 0, stream>>>(hist, W1, b1, hWS);
    teavar_gemm2<<<1, 96, 0, stream>>>(hWS, W2, b2, base_cap, thetaOut, capWS);
    teavar_lp   <<<BD, 256, 0, stream>>>(tmWS, capWS, ebWS, aliveT, edgeT, prob, xOut);
}